// SMFNet_24240795419182
// MI455X (gfx1250) — compile-verified
//
#include <hip/hip_runtime.h>
#include <hip/hip_bf16.h>

// Problem constants (fixed by the reference).
constexpr int N_ROWS = 8192;   // N
constexpr int D_IN   = 128;
constexpr int H1     = 64;
constexpr int NLAYER = 3;

typedef __attribute__((ext_vector_type(2))) float v2f;
typedef __attribute__((ext_vector_type(8))) float v8f;

// ---------------------------------------------------------------------------
// Kernel 1: V0 = relu(X @ g_W + g_b)   [8192x128] x [128x64] -> [8192x64]
// One wave per 16x16 output tile, K advanced 4 per v_wmma_f32_16x16x4_f32.
// A frag (16x4 f32, 2 VGPRs): lanes 0-15 hold K=0,1; lanes 16-31 hold K=2,3.
// B frag (4x16) mirrored; C/D: VGPR r holds M=r (lanes 0-15) / M=r+8 (16-31).
// ---------------------------------------------------------------------------
__global__ void __launch_bounds__(256) gemm_v0_kernel(
    const float* __restrict__ X, const float* __restrict__ gW,
    const float* __restrict__ gb, float* __restrict__ V0)
{
    const int lane = threadIdx.x & 31;
    const int wave = threadIdx.x >> 5;
    const int tile = blockIdx.x * 8 + wave;        // 0..2047
    const int tileM = (tile >> 2) * 16;            // 512 M-tiles
    const int tileN = (tile & 3) * 16;             // 4  N-tiles
    const int hi = lane >> 4;                      // 0/1 lane half
    const int lo = lane & 15;
    const int kb = hi * 2;                         // K sub-offset for this half

    const float* arow = X + (size_t)(tileM + lo) * D_IN;

    v8f c = {};
    for (int k0 = 0; k0 < D_IN; k0 += 4) {
        v2f a, b;
        a.x = arow[k0 + kb + 0];
        a.y = arow[k0 + kb + 1];
        const float* bp = gW + (size_t)(k0 + kb) * H1 + tileN + lo;
        b.x = bp[0];
        b.y = bp[H1];
        // 8 args: (neg_a, A, neg_b, B, c_mod, C, reuse_a, reuse_b)
        c = __builtin_amdgcn_wmma_f32_16x16x4_f32(
                false, a, false, b, (short)0, c, false, false);
    }

    const int col = tileN + lo;
    const float bias = gb[col];
#pragma unroll
    for (int r = 0; r < 8; ++r) {
        const int row = tileM + r + 8 * hi;
        const float v = c[r] + bias;
        V0[row * H1 + col] = v > 0.0f ? v : 0.0f;
    }
}

// ---------------------------------------------------------------------------
// Kernel 2 (per layer): exploit chord sparsity.
//   a_i = relu(X[i]·fW[:,i]   + fb[i])
//   b_i = relu(X[i]·fW[:,i+1] + fb[i+1])      (index mod 8192)
//   Vout[i,:] = a_i*Vin[i,:] + b_i*Vin[i+1,:]
// Thread t <-> row i = bid*256+t : fW column loads are unit-stride across
// lanes (coalesced); X row elements are served from L0 across the k-loop.
// ---------------------------------------------------------------------------
__global__ void __launch_bounds__(256) chord_layer_kernel(
    const float* __restrict__ X, const float* __restrict__ fW,
    const float* __restrict__ fb, const float* __restrict__ Vin,
    float* __restrict__ Vout)
{
    __shared__ float sa[256];
    __shared__ float sb[256];

    const int t  = threadIdx.x;
    const int i  = blockIdx.x * 256 + t;
    const int i1 = (i + 1) & (N_ROWS - 1);

    const float* xr = X + (size_t)i * D_IN;
    float acc0 = 0.0f, acc1 = 0.0f;

    for (int k = 0; k < D_IN; k += 4) {
        if (k + 8 < D_IN) {
            // speculative stream-ahead of the weight matrix (global_prefetch_b8)
            __builtin_prefetch(fW + (size_t)(k + 8) * N_ROWS + i, 0, 1);
        }
        const float4 xv = *reinterpret_cast<const float4*>(xr + k);
        const float* w = fW + (size_t)k * N_ROWS;
        acc0 = fmaf(xv.x, w[i],               acc0);
        acc1 = fmaf(xv.x, w[i1],              acc1);
        acc0 = fmaf(xv.y, w[N_ROWS + i],      acc0);
        acc1 = fmaf(xv.y, w[N_ROWS + i1],     acc1);
        acc0 = fmaf(xv.z, w[2 * N_ROWS + i],  acc0);
        acc1 = fmaf(xv.z, w[2 * N_ROWS + i1], acc1);
        acc0 = fmaf(xv.w, w[3 * N_ROWS + i],  acc0);
        acc1 = fmaf(xv.w, w[3 * N_ROWS + i1], acc1);
    }

    const float av = acc0 + fb[i];
    const float bv = acc1 + fb[i1];
    sa[t] = av > 0.0f ? av : 0.0f;
    sb[t] = bv > 0.0f ? bv : 0.0f;
    __syncthreads();

    // V update: re-parallelize over (row, col) so V traffic is coalesced.
    const int rowBase = blockIdx.x * 256;
#pragma unroll 4
    for (int j = 0; j < 64; ++j) {
        const int idx = t + 256 * j;          // 0..16383
        const int r   = idx >> 6;             // local row 0..255
        const int cix = idx & 63;             // col 0..63
        const int gr  = rowBase + r;
        const int gr1 = (gr + 1) & (N_ROWS - 1);
        Vout[gr * H1 + cix] =
            sa[r] * Vin[gr * H1 + cix] + sb[r] * Vin[gr1 * H1 + cix];
    }
}

extern "C" void kernel_launch(void* const* d_in, const int* in_sizes, int n_in,
                              void* d_out, int out_size, void* d_ws, size_t ws_size,
                              hipStream_t stream) {
    (void)in_sizes; (void)n_in; (void)out_size; (void)ws_size;

    const float* X   = (const float*)d_in[0];   // [8192,128]
    const float* gW  = (const float*)d_in[1];   // [128,64]
    const float* gb  = (const float*)d_in[2];   // [64]
    const float* fW  = (const float*)d_in[3];   // [3,128,8192]
    const float* fb  = (const float*)d_in[4];   // [3,8192]
    // d_in[5] = n_layer (==3, fixed by the reference constants)

    float* out = (float*)d_out;                 // [8192,64]
    float* V0  = (float*)d_ws;                  // ping
    float* V1  = V0 + (size_t)N_ROWS * H1;      // pong

    // V0 = relu(X @ gW + gb): 2048 wave-tiles / 8 waves per block
    gemm_v0_kernel<<<256, 256, 0, stream>>>(X, gW, gb, V0);

    // Three chord layers, ping-pong, last writes d_out.
    const size_t wStride = (size_t)D_IN * N_ROWS;   // elements per layer of f_W
    float* vbuf[2] = {V0, V1};
    for (int m = 0; m < NLAYER; ++m) {
        const float* vin  = vbuf[m & 1];
        float*       vout = (m == NLAYER - 1) ? out : vbuf[(m + 1) & 1];
        chord_layer_kernel<<<N_ROWS / 256, 256, 0, stream>>>(
            X, fW + (size_t)m * wStride, fb + (size_t)m * N_ROWS, vin, vout);
    }
}